// Subtraction2_59966333386862
// MI455X (gfx1250) — compile-verified
//
#include <hip/hip_runtime.h>
#include <cstdint>

// Problem constants (from reference)
#define NN 8
#define CC 64
#define HH 56
#define WW 56
#define KK 7
#define PADV 3
#define PH  (HH + 2 * PADV)   // 62 padded rows/cols
#define LSTRIDE 64            // LDS row stride in floats (16B-aligned rows)
#define QPR (WW / 4)          // 14 float4 quads per row
#define NQ  (HH * WW / 4)     // 784 quads per plane
#define PLANE (HH * WW)       // 3136

typedef float v4f __attribute__((ext_vector_type(4)));

// ---- CDNA5 async global->LDS fill path (compile-time probed) ---------------
#if defined(__HIP_DEVICE_COMPILE__)
#if __has_builtin(__builtin_amdgcn_global_load_async_to_lds_b32)
#define USE_ASYNC_LDS 1
#endif
#endif
#ifndef USE_ASYNC_LDS
#define USE_ASYNC_LDS 0
#endif

#if USE_ASYNC_LDS
typedef __attribute__((address_space(1))) int GI;  // global (AS1) int
typedef __attribute__((address_space(3))) int LI;  // LDS (AS3) int
#endif

__device__ __forceinline__ int reflect_idx(int i) {
  // numpy 'reflect' (no edge repeat): -1 -> 1, 56 -> 54
  i = (i < 0) ? -i : i;
  i = (i >= HH) ? (2 * HH - 2 - i) : i;
  return i;
}

__global__ __launch_bounds__(256) void sub2_refpad_kernel(
    const float* __restrict__ x1, const float* __restrict__ x2,
    float* __restrict__ out) {
  __shared__ __align__(16) float tile[PH * LSTRIDE];  // 15,872 B

  const unsigned tid = threadIdx.x;
  const unsigned nc  = blockIdx.x;   // 0..511  (n*C + c)
  const unsigned kh  = blockIdx.y;   // 0..6

  const float* __restrict__ x2p = x2 + (size_t)nc * PLANE;

  // ---- Fill reflect-padded x2 plane into LDS -------------------------------
  for (unsigned i = tid; i < (unsigned)(PH * PH); i += 256u) {
    const unsigned pr = i / PH;
    const unsigned pc = i - pr * PH;
    const int sh = reflect_idx((int)pr - PADV);
    const int sw = reflect_idx((int)pc - PADV);
#if USE_ASYNC_LDS
    __builtin_amdgcn_global_load_async_to_lds_b32(
        (GI*)(uintptr_t)(x2p + sh * WW + sw),
        (LI*)(uint32_t)(uintptr_t)&tile[pr * LSTRIDE + pc], 0, 0);
#else
    tile[pr * LSTRIDE + pc] = x2p[sh * WW + sw];
#endif
  }
#if USE_ASYNC_LDS
#if __has_builtin(__builtin_amdgcn_s_wait_asynccnt)
  __builtin_amdgcn_s_wait_asynccnt(0);
#else
  asm volatile("s_wait_asynccnt 0" ::: "memory");
#endif
#endif
  __syncthreads();

  // ---- Stream outputs: per quad, 3 aligned LDS b128 loads -> 7 NT b128 stores
  const v4f* __restrict__ x1q = (const v4f*)(x1 + (size_t)nc * PLANE);
  v4f* __restrict__ outq = (v4f*)out;
  const size_t obase = ((size_t)nc * (KK * KK) + (size_t)kh * KK) * NQ;

  for (unsigned q = tid; q < (unsigned)NQ; q += 256u) {
    const unsigned row = q / QPR;             // output h
    const unsigned wq  = (q - row * QPR) * 4; // output w base (multiple of 4)

    const v4f a = x1q[q];

    const float* lrow = &tile[(row + kh) * LSTRIDE + wq];
    __align__(16) float win[12];
    *(v4f*)&win[0] = *(const v4f*)(lrow);
    *(v4f*)&win[4] = *(const v4f*)(lrow + 4);
    *(v4f*)&win[8] = *(const v4f*)(lrow + 8);

#pragma unroll
    for (int kw = 0; kw < KK; ++kw) {
      v4f o;
      o.x = a.x - win[kw + 0];
      o.y = a.y - win[kw + 1];
      o.z = a.z - win[kw + 2];
      o.w = a.w - win[kw + 3];
      __builtin_nontemporal_store(o, &outq[obase + (size_t)kw * NQ + q]);
    }
  }
}

extern "C" void kernel_launch(void* const* d_in, const int* in_sizes, int n_in,
                              void* d_out, int out_size, void* d_ws, size_t ws_size,
                              hipStream_t stream) {
  (void)in_sizes; (void)n_in; (void)out_size; (void)d_ws; (void)ws_size;
  const float* x1 = (const float*)d_in[0];
  const float* x2 = (const float*)d_in[1];
  float* out = (float*)d_out;

  dim3 grid(NN * CC, KK);  // 512 x 7 = 3584 blocks
  dim3 block(256);         // 8 wave32 per block
  sub2_refpad_kernel<<<grid, block, 0, stream>>>(x1, x2, out);
}